// MinDistanceConvLayer2_66305705116157
// MI455X (gfx1250) — compile-verified
//
#include <hip/hip_runtime.h>
#include <hip/hip_bf16.h>

typedef __attribute__((ext_vector_type(2))) float v2f;
typedef __attribute__((ext_vector_type(8))) float v8f;

#define Hd 96
#define Wd 96
#define HWd (Hd * Wd)          // 9216
#define NTILES (HWd / 16)      // 576 column chunks / row tiles
#define NWAVES 8
#define CHUNKS_PER_WAVE (NTILES / NWAVES)  // 72

// out[ij] = max_xy ( -sqrt(|p_ij - p_xy|^2) - f[xy] )
// dist^2 computed entirely inside V_WMMA_F32_16X16X4_F32:
//   A[M][0..1] = (-2*i_M, -2*j_M),  B[0..1][N] = (x_N, y_N),
//   C[M][N]    = |p_M|^2 + |p_N|^2   ==>  D = dist^2 tile (16x16).
// All operands are integers < 2^24, so the WMMA result is EXACT (>= 0),
// no clamp/NaN-guard needed before sqrt.
__global__ __launch_bounds__(256) void mindist_wmma_kernel(
    const float* __restrict__ f, float* __restrict__ out) {
  __shared__ float fsh[HWd];            // 36 KB: feature map staged once
  __shared__ float red[NWAVES][16];     // cross-wave max reduction

  const int tid = threadIdx.x;

  // Stage f into LDS (coalesced, 36 iterations of 256 lanes).
  for (int p = tid; p < HWd; p += 256) fsh[p] = f[p];
  __syncthreads();

  const int wave   = tid >> 5;
  const int lane   = tid & 31;
  const int lane16 = lane & 15;
  const bool lo    = (lane < 16);
  const int m0     = blockIdx.x * 16;   // 16 output points per block

  // --- A operand (16x4 f32): lanes 0-15 carry row M=lane, K0/K1 = (-2i,-2j);
  //     lanes 16-31 carry K2/K3 which we keep zero.
  v2f a;
  {
    int q = m0 + lane16;
    int i = q / Wd, j = q - i * Wd;
    a.x = lo ? (-2.0f * (float)i) : 0.0f;
    a.y = lo ? (-2.0f * (float)j) : 0.0f;
  }

  // --- |p_M|^2 for this lane's M at each C/D row v
  //     (C/D layout: lanes 0-15 -> M=v, lanes 16-31 -> M=v+8).
  float nrow[8];
  {
    int mbase = lo ? 0 : 8;
#pragma unroll
    for (int v = 0; v < 8; ++v) {
      int q = m0 + mbase + v;
      int i = q / Wd, j = q - i * Wd;
      nrow[v] = (float)(i * i + j * j);
    }
  }

  float macc[8];
#pragma unroll
  for (int v = 0; v < 8; ++v) macc[v] = -3.402823e38f;

  // Each wave scans 72 chunks of 16 xy points.
  for (int t = 0; t < CHUNKS_PER_WAVE; ++t) {
    const int n0 = (wave * CHUNKS_PER_WAVE + t) * 16;
    const int p  = n0 + lane16;          // xy point for this lane's N slot
    int x = p / Wd, y = p - x * Wd;
    float xf = (float)x, yf = (float)y;
    float nx = xf * xf + yf * yf;        // |p_N|^2 (valid in both 16-lane halves)
    float fx = fsh[p];                   // f[xy]   (valid in both 16-lane halves)

    // B operand (4x16 f32). Rows K=2,3 are dead (A has zeros there), so we
    // hedge the layout: lower lanes = (x, y); upper lanes = (y, 0) so that
    // whether VGPR0's upper half is row K=1 or row K=2, the product is right.
    v2f b;
    b.x = lo ? xf : yf;
    b.y = lo ? yf : 0.0f;

    // C = |p_M|^2 + |p_N|^2 per slot -> D = dist^2 (exact).
    v8f c;
#pragma unroll
    for (int v = 0; v < 8; ++v) c[v] = nrow[v] + nx;

    v8f d = __builtin_amdgcn_wmma_f32_16x16x4_f32(
        /*neg_a=*/false, a, /*neg_b=*/false, b,
        /*c_mod=*/(short)0, c, /*reuse_a=*/false, /*reuse_b=*/false);

#pragma unroll
    for (int v = 0; v < 8; ++v) {
      // d[v] is exact and >= 0 (pure integer arithmetic in f32 range).
      float val = -__builtin_amdgcn_sqrtf(d[v]) - fx;  // v_sqrt + pk_add(neg)
      macc[v] = fmaxf(macc[v], val);
    }
  }

  // Reduce over N (the 16 lanes of each half-wave) via butterfly shuffles.
#pragma unroll
  for (int v = 0; v < 8; ++v) {
    float m = macc[v];
    m = fmaxf(m, __shfl_xor(m, 1, 32));
    m = fmaxf(m, __shfl_xor(m, 2, 32));
    m = fmaxf(m, __shfl_xor(m, 4, 32));
    m = fmaxf(m, __shfl_xor(m, 8, 32));
    macc[v] = m;
  }
  if (lane == 0) {
#pragma unroll
    for (int v = 0; v < 8; ++v) red[wave][v] = macc[v];        // rows M=0..7
  } else if (lane == 16) {
#pragma unroll
    for (int v = 0; v < 8; ++v) red[wave][8 + v] = macc[v];    // rows M=8..15
  }
  __syncthreads();

  // Cross-wave max and final store.
  if (tid < 16) {
    float m = red[0][tid];
#pragma unroll
    for (int w = 1; w < NWAVES; ++w) m = fmaxf(m, red[w][tid]);
    out[m0 + tid] = m;
  }
}

extern "C" void kernel_launch(void* const* d_in, const int* in_sizes, int n_in,
                              void* d_out, int out_size, void* d_ws, size_t ws_size,
                              hipStream_t stream) {
  (void)in_sizes; (void)n_in; (void)d_ws; (void)ws_size; (void)out_size;
  const float* f = (const float*)d_in[0];
  float* out = (float*)d_out;
  mindist_wmma_kernel<<<NTILES, 256, 0, stream>>>(f, out);
}